// GIN_1layer_48266842472560
// MI455X (gfx1250) — compile-verified
//
#include <hip/hip_runtime.h>

typedef __attribute__((ext_vector_type(2))) float v2f;
typedef __attribute__((ext_vector_type(4))) float v4f;
typedef __attribute__((ext_vector_type(8))) float v8f;

#define N_NODES 100000
#define N_EDGES 800000
#define D_FEAT  128
#define CLASSES 64

// ---------------------------------------------------------------------------
// Kernel 1: edge scatter. One wave32 per edge. h must be pre-initialized to x
// (fuses the "(1+eps)*x" term with eps=0). Each lane handles 4 consecutive
// floats of the 128-float feature: one b128 load + 4 f32 atomic adds.
// x and h both fit in L2 (51.2 MB each, 192 MB L2), so the random traffic
// stays on-chip.
// ---------------------------------------------------------------------------
__global__ __launch_bounds__(256) void gin_scatter(
    const float* __restrict__ x,
    const long long* __restrict__ edge_index,
    float* __restrict__ h)
{
    const int wave = blockIdx.x * (256 / 32) + (threadIdx.x >> 5);
    const int lane = threadIdx.x & 31;
    if (wave >= N_EDGES) return;

    const long long src = edge_index[wave];
    const long long dst = edge_index[N_EDGES + wave];

    const v4f v = reinterpret_cast<const v4f*>(x + src * D_FEAT)[lane];
    float* hp = h + dst * D_FEAT + lane * 4;

    __hip_atomic_fetch_add(hp + 0, v.x, __ATOMIC_RELAXED, __HIP_MEMORY_SCOPE_AGENT);
    __hip_atomic_fetch_add(hp + 1, v.y, __ATOMIC_RELAXED, __HIP_MEMORY_SCOPE_AGENT);
    __hip_atomic_fetch_add(hp + 2, v.z, __ATOMIC_RELAXED, __HIP_MEMORY_SCOPE_AGENT);
    __hip_atomic_fetch_add(hp + 3, v.w, __ATOMIC_RELAXED, __HIP_MEMORY_SCOPE_AGENT);
}

// ---------------------------------------------------------------------------
// Kernel 2: out = h @ W + b using V_WMMA_F32_16X16X4_F32 (f32-exact).
// One wave computes one 16x16 output tile; K=128 in 32 steps of 4.
//
// A 16x4 f32 frag (2 VGPRs): lane L holds row M=L&15; VGPR0=K base, VGPR1=K+1,
//   with lanes 16-31 covering K+2/K+3 -> each lane loads 2 consecutive floats.
// B 4x16 frag (2 VGPRs): lane L holds col N=L&15; lanes 0-15 rows K,K+1,
//   lanes 16-31 rows K+2,K+3.
// C/D 16x16 f32 (8 VGPRs): VGPR i holds M = i + 8*(lane>=16), N = lane&15.
// ---------------------------------------------------------------------------
__global__ __launch_bounds__(128) void gin_gemm(
    const float* __restrict__ h,
    const float* __restrict__ W,
    const float* __restrict__ bias,
    float* __restrict__ out)
{
    const int wave = blockIdx.x * 4 + (threadIdx.x >> 5);
    const int lane = threadIdx.x & 31;

    const int rowTile = wave >> 2;          // 6250 row tiles
    const int colTile = wave & 3;           // 4 col tiles of 16
    if (rowTile >= N_NODES / 16) return;

    const int row0  = rowTile * 16;
    const int col0  = colTile * 16;
    const int mn    = lane & 15;            // M for A, N for B/C/D
    const int khalf = (lane >> 4) * 2;      // 0 or 2: which K-pair this lane holds

    const float* arow = h + (long long)(row0 + mn) * D_FEAT + khalf;
    const float* bcol = W + col0 + mn;

    v8f acc = {};
#pragma unroll
    for (int k = 0; k < D_FEAT; k += 4) {
        v2f a;
        a.x = arow[k];                              // consecutive -> b64 load
        a.y = arow[k + 1];
        v2f bb;
        bb.x = bcol[(k + khalf) * CLASSES];
        bb.y = bcol[(k + khalf + 1) * CLASSES];
        acc = __builtin_amdgcn_wmma_f32_16x16x4_f32(
            /*neg_a=*/false, a, /*neg_b=*/false, bb,
            /*c_mod=*/(short)0, acc, /*reuse_a=*/false, /*reuse_b=*/false);
    }

    const float bv = bias[col0 + mn];
    const int mbase = row0 + (lane >> 4) * 8;
#pragma unroll
    for (int i = 0; i < 8; ++i) {
        out[(long long)(mbase + i) * CLASSES + col0 + mn] = acc[i] + bv;
    }
}

// ---------------------------------------------------------------------------
// Launch: h(=ws) <- x (async d2d copy), scatter edges, then WMMA GEMM.
// ws needs N_NODES*D_FEAT*4 = 51.2 MB.
// ---------------------------------------------------------------------------
extern "C" void kernel_launch(void* const* d_in, const int* in_sizes, int n_in,
                              void* d_out, int out_size, void* d_ws, size_t ws_size,
                              hipStream_t stream)
{
    const float*      x  = (const float*)d_in[0];
    const long long*  ei = (const long long*)d_in[1];
    const float*      W  = (const float*)d_in[2];
    const float*      b  = (const float*)d_in[3];
    float*            out = (float*)d_out;
    float*            h   = (float*)d_ws;

    hipMemcpyAsync(h, x, (size_t)N_NODES * D_FEAT * sizeof(float),
                   hipMemcpyDeviceToDevice, stream);

    // 800000 edges, 8 waves (edges) per 256-thread block -> 100000 blocks
    gin_scatter<<<N_EDGES / 8, 256, 0, stream>>>(x, ei, h);

    // 6250*4 = 25000 tile-waves, 4 waves per 128-thread block -> 6250 blocks
    gin_gemm<<<(N_NODES / 16) * 4 / 4, 128, 0, stream>>>(h, W, b, out);
}